// CTM_15272903704828
// MI455X (gfx1250) — compile-verified
//
#include <hip/hip_runtime.h>
#include <math.h>

// ---- problem constants (match reference) ----
#define BB   4
#define NN   5120
#define CC   256
#define MM   128
#define KNN  9
#define RT   16          // WMMA tile edge
#define NT   (NN / RT)   // 320 column tiles
#define RB   64          // rows per workgroup in the N^2 passes
#define RWAVES 4         // 128-thread block = 4 wave32, one row-tile each
#define WAVES 8          // 256-thread block for k5

typedef __attribute__((ext_vector_type(2))) float v2f;
typedef __attribute__((ext_vector_type(8))) float v8f;

// One wave computes a 16x16 Gram tile  G = Arows · Bcols^T  over K = CC
// using V_WMMA_F32_16X16X4_F32 (64 chained WMMAs).
// A 16x4 layout (ISA 7.12.2): lanes 0-15 hold M=0..15 with K=k,k+1 in v0,v1;
// lanes 16-31 hold K=k+2,k+3.  B mirrors with N=lane&15.
__device__ __forceinline__ v8f gram_tile(const float* __restrict__ aRow, // LDS [16][CC]
                                         const float* __restrict__ bCol, // 16 rows of CC
                                         int lane) {
  const int half = lane >> 4;
  const int l16  = lane & 15;
  const float* ap = aRow + l16 * CC + half * 2;
  const float* bp = bCol + (size_t)l16 * CC + half * 2;
  v8f acc = {0.f, 0.f, 0.f, 0.f, 0.f, 0.f, 0.f, 0.f};
#pragma unroll 8
  for (int k = 0; k < CC; k += 4) {
    v2f a; a.x = ap[k]; a.y = ap[k + 1];
    v2f b; b.x = bp[k]; b.y = bp[k + 1];
    acc = __builtin_amdgcn_wmma_f32_16x16x4_f32(false, a, false, b,
                                                (short)0, acc, false, false);
  }
  return acc;
}

// Wave 0 prefetches the next column tile (16 rows x 1KB) into near caches:
// lane covers half a row = 512B = 4 x 128B lines.  -> global_prefetch_b8
__device__ __forceinline__ void prefetch_tile(const float* __restrict__ xb,
                                              int ctNext, int half, int l16) {
  const float* pl = xb + ((size_t)ctNext * RT + l16) * CC + half * (CC / 2);
#pragma unroll
  for (int j = 0; j < 4; ++j) __builtin_prefetch(pl + j * 32, 0, 3);
}

// K1: sq norms + token weights; zero per-batch d2max.
__global__ __launch_bounds__(256) void k1_sq_tw(const float* __restrict__ x,
                                                const float* __restrict__ sw,
                                                const float* __restrict__ sb,
                                                float* __restrict__ sq,
                                                float* __restrict__ tw,
                                                unsigned* __restrict__ d2max) {
  int gid = blockIdx.x * blockDim.x + threadIdx.x;
  if (gid < BB) d2max[gid] = 0u;
  int tok  = gid >> 5;  // one wave per token
  int lane = threadIdx.x & 31;
  if (tok >= BB * NN) return;
  const float* xp = x + (size_t)tok * CC;
  float s = 0.f, d = 0.f;
  for (int c = lane; c < CC; c += 32) { float v = xp[c]; s += v * v; d += v * sw[c]; }
  for (int m = 16; m; m >>= 1) { s += __shfl_xor(s, m, 32); d += __shfl_xor(d, m, 32); }
  if (lane == 0) { sq[tok] = s; tw[tok] = expf(d + sb[0]); }
}

// K2: 64-row block; fused Gram -> d2 -> per-row streaming top-9 -> density.
// Each wave owns 16 rows; all 4 waves sweep the same column tile (B hits WGP$).
__global__ __launch_bounds__(128) void k2_density(const float* __restrict__ x,
                                                  const float* __restrict__ sq,
                                                  float* __restrict__ density,
                                                  unsigned* __restrict__ d2max) {
  __shared__ float rowX[RB * CC];               // 64 KB
  __shared__ float rowSq[RB];
  __shared__ float tileBuf[RWAVES * RT * RT];   // wave-private 16x16 staging
  const int b = blockIdx.y;
  const int rowbase = blockIdx.x * RB;
  const float* xb = x + (size_t)b * NN * CC;
  for (int i = threadIdx.x; i < RB * CC; i += 128) rowX[i] = xb[(size_t)rowbase * CC + i];
  if (threadIdx.x < RB) rowSq[threadIdx.x] = sq[b * NN + rowbase + threadIdx.x];
  __syncthreads();

  const int wave = threadIdx.x >> 5, lane = threadIdx.x & 31;
  const int half = lane >> 4, l16 = lane & 15;
  float* myTile = tileBuf + wave * (RT * RT);
  const float* aRow = rowX + wave * RT * CC;
  const float* mySq = rowSq + wave * RT;

  float tk[KNN];
#pragma unroll
  for (int i = 0; i < KNN; ++i) tk[i] = 3.4e38f;
  float md2 = 0.f;

  for (int ct = 0; ct < NT; ++ct) {
    if (wave == 0) prefetch_tile(xb, (ct + 1 < NT) ? ct + 1 : ct, half, l16);
    const int colbase = ct * RT;
    v8f g = gram_tile(aRow, xb + (size_t)colbase * CC, lane);
    const float sqc = sq[b * NN + colbase + l16];
#pragma unroll
    for (int r = 0; r < 8; ++r) {
      float d2 = fmaxf(mySq[half * 8 + r] + sqc - 2.0f * g[r], 0.f);
      md2 = fmaxf(md2, d2);
      myTile[(half * 8 + r) * RT + l16] = d2;
    }
    // wave-private staging: LDS ops from one wave execute in order (ISA);
    // only compiler reordering must be fenced.
    __builtin_amdgcn_wave_barrier();
    if (lane < RT) {                 // lane owns row `lane` of this wave's tile
      const float* rowv = myTile + lane * RT;
#pragma unroll
      for (int c2 = 0; c2 < RT; ++c2) {
        float v = rowv[c2];
        if (v < tk[KNN - 1]) {       // keep ascending-sorted top-9 smallest
          tk[KNN - 1] = v;
#pragma unroll
          for (int i = KNN - 1; i > 0; --i)
            if (tk[i] < tk[i - 1]) { float t = tk[i - 1]; tk[i - 1] = tk[i]; tk[i] = t; }
        }
      }
    }
    __builtin_amdgcn_wave_barrier();
  }
  if (lane < RT) {
    float s = 0.f;
#pragma unroll
    for (int i = 0; i < KNN; ++i) s += tk[i];
    // dist^2 = d2 / C ; density = exp(-mean of 9 smallest dist^2)
    density[b * NN + rowbase + wave * RT + lane] = expf(-s * (1.0f / (KNN * (float)CC)));
  }
  for (int m = 16; m; m >>= 1) md2 = fmaxf(md2, __shfl_xor(md2, m, 32));
  if (lane == 0) atomicMax(&d2max[b], __float_as_uint(md2)); // nonneg float bits
}

// K3: 64-row block; fused Gram -> dist -> masked running-min -> score.
// Barrier-free main loop: per-lane running min, one butterfly at the end.
__global__ __launch_bounds__(128) void k3_score(const float* __restrict__ x,
                                                const float* __restrict__ sq,
                                                const float* __restrict__ density,
                                                const unsigned* __restrict__ d2max,
                                                float* __restrict__ score) {
  __shared__ float rowX[RB * CC];
  __shared__ float rowSq[RB];
  __shared__ float rowDen[RB];
  const int b = blockIdx.y;
  const int rowbase = blockIdx.x * RB;
  const float* xb = x + (size_t)b * NN * CC;
  for (int i = threadIdx.x; i < RB * CC; i += 128) rowX[i] = xb[(size_t)rowbase * CC + i];
  if (threadIdx.x < RB) {
    rowSq[threadIdx.x]  = sq[b * NN + rowbase + threadIdx.x];
    rowDen[threadIdx.x] = density[b * NN + rowbase + threadIdx.x];
  }
  __syncthreads();

  const int wave = threadIdx.x >> 5, lane = threadIdx.x & 31;
  const int half = lane >> 4, l16 = lane & 15;
  const float* aRow = rowX + wave * RT * CC;
  const float dmax = sqrtf(__uint_as_float(d2max[b])) * 0.0625f; // /sqrt(C)

  float minAcc[8];
#pragma unroll
  for (int r = 0; r < 8; ++r) minAcc[r] = 3.4e38f;

  for (int ct = 0; ct < NT; ++ct) {
    if (wave == 0) prefetch_tile(xb, (ct + 1 < NT) ? ct + 1 : ct, half, l16);
    const int colbase = ct * RT;
    v8f g = gram_tile(aRow, xb + (size_t)colbase * CC, lane);
    const float sqc  = sq[b * NN + colbase + l16];
    const float denc = density[b * NN + colbase + l16];
#pragma unroll
    for (int r = 0; r < 8; ++r) {
      float d2   = fmaxf(rowDen[0] * 0.f + rowSq[wave * RT + half * 8 + r] + sqc - 2.0f * g[r], 0.f);
      float dist = sqrtf(d2) * 0.0625f;
      float cand = (denc > rowDen[wave * RT + half * 8 + r]) ? dist : dmax;
      minAcc[r] = fminf(minAcc[r], cand);
    }
  }
#pragma unroll
  for (int r = 0; r < 8; ++r) {
    float v = minAcc[r];
    for (int m2 = 1; m2 < 16; m2 <<= 1) v = fminf(v, __shfl_xor(v, m2, 32));
    if (l16 == 0) {
      const int rloc = wave * RT + half * 8 + r;
      score[b * NN + rowbase + rloc] = v * rowDen[rloc];
    }
  }
}

// K4: per-batch iterative top-128 + gather center features.
__global__ __launch_bounds__(256) void k4_select(const float* __restrict__ x,
                                                 const float* __restrict__ sq,
                                                 const float* __restrict__ score,
                                                 int* __restrict__ index_down,
                                                 float* __restrict__ centers,
                                                 float* __restrict__ centerSq) {
  __shared__ float sc[NN];
  __shared__ unsigned long long red[256];
  __shared__ int curIdx;
  const int b = blockIdx.x;
  for (int i = threadIdx.x; i < NN; i += 256) sc[i] = score[b * NN + i];
  __syncthreads();
  for (int m = 0; m < MM; ++m) {
    unsigned long long best = 0ull;
    for (int i = threadIdx.x; i < NN; i += 256) {
      float v = sc[i];
      if (v >= 0.f) {   // scores are nonneg; selected entries set to -1
        unsigned long long key =
            ((unsigned long long)__float_as_uint(v) << 32) | (unsigned)(NN - 1 - i);
        best = key > best ? key : best;  // tie -> lower index (jax top_k)
      }
    }
    red[threadIdx.x] = best;
    __syncthreads();
    for (int s = 128; s; s >>= 1) {
      if (threadIdx.x < s && red[threadIdx.x + s] > red[threadIdx.x])
        red[threadIdx.x] = red[threadIdx.x + s];
      __syncthreads();
    }
    if (threadIdx.x == 0) {
      int idx = NN - 1 - (int)(red[0] & 0xffffffffu);
      curIdx = idx;
      index_down[b * MM + m] = idx;
      sc[idx] = -1.0f;
    }
    __syncthreads();
    int idx = curIdx;
    for (int c = threadIdx.x; c < CC; c += 256)
      centers[((size_t)b * MM + m) * CC + c] = x[((size_t)b * NN + idx) * CC + c];
    if (threadIdx.x == 0) centerSq[b * MM + m] = sq[b * NN + idx];
    __syncthreads();
  }
}

// K5: WMMA token x center distances; per-token argmin over 128 centers.
__global__ __launch_bounds__(256) void k5_assign(const float* __restrict__ x,
                                                 const float* __restrict__ sq,
                                                 const float* __restrict__ centers,
                                                 const float* __restrict__ centerSq,
                                                 int* __restrict__ idxc) {
  __shared__ float rowX[RT * CC];
  __shared__ float rowSq[RT];
  __shared__ unsigned long long best[RT];
  const int b = blockIdx.y;
  const int rowbase = blockIdx.x * RT;
  const float* xb = x + (size_t)b * NN * CC;
  for (int i = threadIdx.x; i < RT * CC; i += 256) rowX[i] = xb[(size_t)rowbase * CC + i];
  if (threadIdx.x < RT) {
    rowSq[threadIdx.x] = sq[b * NN + rowbase + threadIdx.x];
    best[threadIdx.x]  = ~0ull;
  }
  __syncthreads();
  const int wave = threadIdx.x >> 5, lane = threadIdx.x & 31;
  const int half = lane >> 4, l16 = lane & 15;
  const int col = wave * 16 + l16;  // 8 waves cover all 128 centers
  v8f g = gram_tile(rowX, centers + ((size_t)b * MM + wave * 16) * CC, lane);
  const float sqc = centerSq[b * MM + col];
#pragma unroll
  for (int r = 0; r < 8; ++r) {
    float d2 = fmaxf(rowSq[half * 8 + r] + sqc - 2.0f * g[r], 0.f);
    unsigned long long key =
        ((unsigned long long)__float_as_uint(d2) << 32) | (unsigned)col;
    for (int m2 = 1; m2 < 16; m2 <<= 1) {
      unsigned long long o = __shfl_xor(key, m2, 32);
      key = o < key ? o : key;   // tie -> lower center index (argmin)
    }
    if (l16 == 0) atomicMin(&best[half * 8 + r], key);
  }
  __syncthreads();
  if (threadIdx.x < RT)
    idxc[b * NN + rowbase + threadIdx.x] = (int)(best[threadIdx.x] & 0xffffffffu);
}

// K6: zero out/all_w (replay-safe) + override centers' own cluster index.
__global__ __launch_bounds__(256) void k6_fixup(const int* __restrict__ index_down,
                                                int* __restrict__ idxc,
                                                float* __restrict__ all_w,
                                                float* __restrict__ out) {
  int gid = blockIdx.x * blockDim.x + threadIdx.x;
  if (gid < BB * MM * CC) out[gid] = 0.f;
  if (gid < BB * MM) {
    all_w[gid] = 0.f;
    int bq = gid / MM, mq = gid % MM;
    idxc[bq * NN + index_down[gid]] = mq;
  }
}

// K7: segment-sum of weights.
__global__ __launch_bounds__(256) void k7_allw(const int* __restrict__ idxc,
                                               const float* __restrict__ tw,
                                               float* __restrict__ all_w) {
  int gid = blockIdx.x * blockDim.x + threadIdx.x;
  if (gid >= BB * NN) return;
  int b = gid / NN;
  atomicAdd(&all_w[b * MM + idxc[gid]], tw[gid]);
}

// K8: weighted merge: out[b,m,:] += x[b,n,:] * w_n / (all_w + 1e-6).
__global__ __launch_bounds__(256) void k8_merge(const float* __restrict__ x,
                                                const int* __restrict__ idxc,
                                                const float* __restrict__ tw,
                                                const float* __restrict__ all_w,
                                                float* __restrict__ out) {
  const int t = blockIdx.x;          // token id 0..B*N-1
  const int b = t / NN;
  const int m = idxc[t];
  const float nw = tw[t] / (all_w[b * MM + m] + 1e-6f);
  const float* xp = x + (size_t)t * CC;
  float* op = out + ((size_t)b * MM + m) * CC;
  for (int c = threadIdx.x; c < CC; c += 256) atomicAdd(&op[c], xp[c] * nw);
}

extern "C" void kernel_launch(void* const* d_in, const int* in_sizes, int n_in,
                              void* d_out, int out_size, void* d_ws, size_t ws_size,
                              hipStream_t stream) {
  const float* x  = (const float*)d_in[0];   // [B,N,C]
  const float* sw = (const float*)d_in[1];   // [C,1]
  const float* sb = (const float*)d_in[2];   // [1]
  float* out = (float*)d_out;                // [B,M,C]

  // workspace layout (floats / ints, ~0.94 MB total)
  float* w = (float*)d_ws;
  float* sq       = w;                 w += BB * NN;
  float* tw       = w;                 w += BB * NN;
  float* density  = w;                 w += BB * NN;
  float* score    = w;                 w += BB * NN;
  float* centers  = w;                 w += BB * MM * CC;
  float* centerSq = w;                 w += BB * MM;
  float* all_w    = w;                 w += BB * MM;
  unsigned* d2max = (unsigned*)w;      w += BB;
  int* index_down = (int*)w;           w += BB * MM;
  int* idxc       = (int*)w;           w += BB * NN;

  k1_sq_tw<<<(BB * NN * 32) / 256, 256, 0, stream>>>(x, sw, sb, sq, tw, d2max);
  k2_density<<<dim3(NN / RB, BB), 128, 0, stream>>>(x, sq, density, d2max);
  k3_score<<<dim3(NN / RB, BB), 128, 0, stream>>>(x, sq, density, d2max, score);
  k4_select<<<BB, 256, 0, stream>>>(x, sq, score, index_down, centers, centerSq);
  k5_assign<<<dim3(NT, BB), 256, 0, stream>>>(x, sq, centers, centerSq, idxc);
  k6_fixup<<<(BB * MM * CC + 255) / 256, 256, 0, stream>>>(index_down, idxc, all_w, out);
  k7_allw<<<(BB * NN + 255) / 256, 256, 0, stream>>>(idxc, tw, all_w);
  k8_merge<<<BB * NN, 256, 0, stream>>>(x, idxc, tw, all_w, out);
}